// GCNModel0_78176994722446
// MI455X (gfx1250) — compile-verified
//
#include <hip/hip_runtime.h>
#include <math.h>

#define N_NODES 100000
#define N_EDGES 1600000
#define N_GRAPHS 256
#define F_IN 7
#define H 64
#define WPAD 80   // float2 stride per K-row-pair in LDS (160 dwords = 32 mod 64 banks)

typedef __attribute__((ext_vector_type(2))) float v2f;
typedef __attribute__((ext_vector_type(8))) float v8f;

// ---------------- utility ----------------
__global__ void zero_kernel(float* __restrict__ p, int n) {
    int i = blockIdx.x * blockDim.x + threadIdx.x;
    if (i < n) p[i] = 0.0f;
}

// deg[dst] += 1 over all edges (stored as float so we can rsqrt in place)
__global__ void deg_kernel(const int* __restrict__ dst, float* __restrict__ deg, int e) {
    int i = blockIdx.x * blockDim.x + threadIdx.x;
    if (i < e) atomicAdd(&deg[dst[i]], 1.0f);
}

// dinv[n] = rsqrt(deg[n] + 1)   (self-loop included, deg >= 1 always)
__global__ void dinv_kernel(float* __restrict__ d, int n) {
    int i = blockIdx.x * blockDim.x + threadIdx.x;
    if (i < n) d[i] = rsqrtf(d[i] + 1.0f);
}

// ---------------- layer 1: x[N,7] @ W1[7,64], scalar (K=7 too small for WMMA) ----
__global__ void mm7_kernel(const float* __restrict__ x, const float* __restrict__ W,
                           const float* __restrict__ dinv, float* __restrict__ hlin,
                           float* __restrict__ agg, int n) {
    int i = blockIdx.x * blockDim.x + threadIdx.x;
    if (i >= n * H) return;
    int node = i >> 6, f = i & 63;
    float s = 0.0f;
#pragma unroll
    for (int k = 0; k < F_IN; ++k) s += x[node * F_IN + k] * W[k * H + f];
    hlin[i] = s;
    float di = dinv[node];
    agg[i] = s * di * di;   // self-loop term initializes the aggregate
}

// ---------------- layers 2/3: h[N,64] @ W[64,64] via V_WMMA_F32_16X16X4_F32 ------
// One wave32 per 16-row node tile. W staged in LDS with K-row pairs interleaved:
// Wl[p*WPAD + col] = {W[2p][col], W[2p+1][col]}  -> each B-fragment is a single
// contiguous ds_load_b64 into an even-aligned VGPR pair (no pack movs), and the
// WPAD stride puts the two half-waves on disjoint bank ranges.
__global__ __launch_bounds__(256) void mm_wmma_kernel(
        const float* __restrict__ hin, const float* __restrict__ W,
        const float* __restrict__ dinv, float* __restrict__ hlin,
        float* __restrict__ agg, int ntiles) {
    __shared__ v2f Wl[(H / 2) * WPAD];
    for (int p = threadIdx.x; p < (H / 2) * H; p += 256) {
        int kp = p >> 6;         // K-row pair index 0..31
        int col = p & 63;
        v2f w;
        w.x = W[(kp * 2) * H + col];
        w.y = W[(kp * 2 + 1) * H + col];
        Wl[kp * WPAD + col] = w;
    }
    __syncthreads();

    int wave = threadIdx.x >> 5;
    int lane = threadIdx.x & 31;
    int tile = blockIdx.x * 8 + wave;
    if (tile >= ntiles) return;           // wave-uniform: EXEC stays all-ones

    int m = lane & 15;
    int khalf = lane >> 4;                // 0: K=k..k+1 rows, 1: K=k+2..k+3
    int rowA = tile * 16 + m;

    // Preload full A tile: 16 k-steps, float2 per step.
    v2f a[16];
    const float* hrow = hin + (size_t)rowA * H;
#pragma unroll
    for (int kk = 0; kk < 16; ++kk) {
        int kb = kk * 4 + khalf * 2;
        a[kk] = *(const v2f*)(hrow + kb);
    }

#pragma unroll
    for (int t = 0; t < 4; ++t) {         // 4 output column tiles of 16
        int col = t * 16 + m;
        v8f c = {};
#pragma unroll
        for (int kk = 0; kk < 16; ++kk) {
            // K-row pair index for this step = (kk*4 + khalf*2)/2
            v2f b = Wl[(kk * 2 + khalf) * WPAD + col];
            c = __builtin_amdgcn_wmma_f32_16x16x4_f32(
                    false, a[kk], false, b, (short)0, c, false, false);
        }
        // C/D layout: VGPR r -> row base + 8*khalf + r, N = col
#pragma unroll
        for (int r = 0; r < 8; ++r) {
            int rowC = tile * 16 + khalf * 8 + r;
            size_t idx = (size_t)rowC * H + col;
            float v = c[r];
            hlin[idx] = v;
            float di = dinv[rowC];
            agg[idx] = v * di * di;
        }
    }
}

// ---------------- edge scatter: agg[dst] += hlin[src] * dinv[src]*dinv[dst] -----
// 32 lanes per edge, float2 per lane -> 2 f32 atomics per lane, L2-resident reads.
__global__ void edge_kernel(const int* __restrict__ src, const int* __restrict__ dst,
                            const float* __restrict__ hlin, const float* __restrict__ dinv,
                            float* __restrict__ agg, int e) {
    long long gid = (long long)blockIdx.x * blockDim.x + threadIdx.x;
    if (gid >= (long long)e * 32) return;
    int edge = (int)(gid >> 5);
    int fp = (int)(gid & 31);
    int s = src[edge], d = dst[edge];
    float nrm = dinv[s] * dinv[d];
    v2f hv = *(const v2f*)(hlin + (size_t)s * H + fp * 2);
    float* ap = agg + (size_t)d * H + fp * 2;
    atomicAdd(ap, hv.x * nrm);
    atomicAdd(ap + 1, hv.y * nrm);
}

// ---------------- bias + ELU ----------------
__global__ void act_kernel(const float* __restrict__ agg, const float* __restrict__ b,
                           float* __restrict__ hout, int total) {
    int i = blockIdx.x * blockDim.x + threadIdx.x;
    if (i >= total) return;
    float v = agg[i] + b[i & 63];
    hout[i] = v > 0.0f ? v : expm1f(v);
}

// ---------------- mean pool over graphs ----------------
__global__ void pool_kernel(const float* __restrict__ h, const int* __restrict__ batch,
                            float* __restrict__ sums, float* __restrict__ cnt, int n) {
    int i = blockIdx.x * blockDim.x + threadIdx.x;
    if (i >= n * H) return;
    int node = i >> 6, f = i & 63;
    int g = batch[node];
    atomicAdd(&sums[g * H + f], h[i]);
    if (f == 0) atomicAdd(&cnt[g], 1.0f);
}

__global__ void readout_kernel(const float* __restrict__ sums, const float* __restrict__ cnt,
                               float* __restrict__ ro, int total) {
    int i = blockIdx.x * blockDim.x + threadIdx.x;
    if (i >= total) return;
    float c = cnt[i >> 6];
    ro[i] = sums[i] / (c < 1.0f ? 1.0f : c);
}

// ---------------- MLP head: one block (64 threads) per graph ----------------
__global__ __launch_bounds__(64) void head_kernel(
        const float* __restrict__ ro,
        const float* __restrict__ Wp1, const float* __restrict__ bp1,
        const float* __restrict__ Wp2, const float* __restrict__ bp2,
        const float* __restrict__ Wp3, const float* __restrict__ bp3,
        float* __restrict__ out) {
    __shared__ float r[H];
    __shared__ float z1[H];
    __shared__ float z2[H / 2];
    int g = blockIdx.x, t = threadIdx.x;
    r[t] = ro[(size_t)g * H + t];
    __syncthreads();
    float s = bp1[t];
    for (int k = 0; k < H; ++k) s += r[k] * Wp1[k * H + t];
    z1[t] = s > 0.0f ? s : 0.0f;
    __syncthreads();
    if (t < H / 2) {
        float s2 = bp2[t];
        for (int k = 0; k < H; ++k) s2 += z1[k] * Wp2[k * (H / 2) + t];
        z2[t] = s2 > 0.0f ? s2 : 0.0f;
    }
    __syncthreads();
    if (t == 0) {
        float s3 = bp3[0];
        for (int k = 0; k < H / 2; ++k) s3 += z2[k] * Wp3[k];
        out[g] = s3;
    }
}

extern "C" void kernel_launch(void* const* d_in, const int* in_sizes, int n_in,
                              void* d_out, int out_size, void* d_ws, size_t ws_size,
                              hipStream_t stream) {
    const float* x    = (const float*)d_in[0];
    const int*   ei   = (const int*)d_in[1];
    const int*   bat  = (const int*)d_in[2];
    const float* W1   = (const float*)d_in[3];
    const float* b1   = (const float*)d_in[4];
    const float* W2   = (const float*)d_in[5];
    const float* b2   = (const float*)d_in[6];
    const float* W3   = (const float*)d_in[7];
    const float* b3   = (const float*)d_in[8];
    const float* Wp1  = (const float*)d_in[9];
    const float* bp1  = (const float*)d_in[10];
    const float* Wp2  = (const float*)d_in[11];
    const float* bp2  = (const float*)d_in[12];
    const float* Wp3  = (const float*)d_in[13];
    const float* bp3  = (const float*)d_in[14];
    const int* src = ei;                 // edge_index[0]
    const int* dst = ei + N_EDGES;       // edge_index[1]

    float* out = (float*)d_out;          // [B]
    float* readout = out + N_GRAPHS;     // [B, H]

    float* ws   = (float*)d_ws;
    float* dinv = ws;                                        // N (also deg scratch)
    float* bufH = ws + ((N_NODES + 255) & ~255);             // N*H  (h_prev / h_next)
    float* bufL = bufH + (size_t)N_NODES * H;                // N*H  (hlin)
    float* bufA = bufL + (size_t)N_NODES * H;                // N*H  (agg)
    float* sums = bufA + (size_t)N_NODES * H;                // B*H
    float* cnt  = sums + N_GRAPHS * H;                       // B

    const int tpb = 256;
    const int nh = N_NODES * H;
    const int nhb = (nh + tpb - 1) / tpb;
    const int eblocks = (int)(((long long)N_EDGES * 32 + tpb - 1) / tpb);
    const int ntiles = N_NODES / 16;
    const int mblocks = (ntiles + 7) / 8;

    // degrees + normalization
    zero_kernel<<<(N_NODES + tpb - 1) / tpb, tpb, 0, stream>>>(dinv, N_NODES);
    deg_kernel<<<(N_EDGES + tpb - 1) / tpb, tpb, 0, stream>>>(dst, dinv, N_EDGES);
    dinv_kernel<<<(N_NODES + tpb - 1) / tpb, tpb, 0, stream>>>(dinv, N_NODES);

    // layer 1 (K=7, scalar matmul)
    mm7_kernel<<<nhb, tpb, 0, stream>>>(x, W1, dinv, bufL, bufA, N_NODES);
    edge_kernel<<<eblocks, tpb, 0, stream>>>(src, dst, bufL, dinv, bufA, N_EDGES);
    act_kernel<<<nhb, tpb, 0, stream>>>(bufA, b1, bufH, nh);

    // layer 2 (WMMA f32)
    mm_wmma_kernel<<<mblocks, tpb, 0, stream>>>(bufH, W2, dinv, bufL, bufA, ntiles);
    edge_kernel<<<eblocks, tpb, 0, stream>>>(src, dst, bufL, dinv, bufA, N_EDGES);
    act_kernel<<<nhb, tpb, 0, stream>>>(bufA, b2, bufH, nh);

    // layer 3 (WMMA f32)
    mm_wmma_kernel<<<mblocks, tpb, 0, stream>>>(bufH, W3, dinv, bufL, bufA, ntiles);
    edge_kernel<<<eblocks, tpb, 0, stream>>>(src, dst, bufL, dinv, bufA, N_EDGES);
    act_kernel<<<nhb, tpb, 0, stream>>>(bufA, b3, bufH, nh);

    // global mean pool
    zero_kernel<<<(N_GRAPHS * H + N_GRAPHS + tpb - 1) / tpb, tpb, 0, stream>>>(
        sums, N_GRAPHS * H + N_GRAPHS);   // sums and cnt are contiguous
    pool_kernel<<<nhb, tpb, 0, stream>>>(bufH, bat, sums, cnt, N_NODES);
    readout_kernel<<<(N_GRAPHS * H + tpb - 1) / tpb, tpb, 0, stream>>>(
        sums, cnt, readout, N_GRAPHS * H);

    // MLP head
    head_kernel<<<N_GRAPHS, 64, 0, stream>>>(readout, Wp1, bp1, Wp2, bp2, Wp3, bp3, out);
}